// MaxKGraphConv_62388694942255
// MI455X (gfx1250) — compile-verified
//
#include <hip/hip_runtime.h>

typedef __attribute__((ext_vector_type(2))) float v2f;
typedef __attribute__((ext_vector_type(4))) float v4f;
typedef __attribute__((ext_vector_type(8))) float v8f;

#define N_NODES   100000
#define N_EDGES   1600000
#define IN_FEATS  256
#define OUT_FEATS 32
#define K_TOP     32
#define KSTEPS    (IN_FEATS / 4)        // 64 WMMA k-steps

// ---------------- kernel 1: zero counters + output accumulator ----------------
__global__ void zero_k(float* a, int na, float* b, int nb) {
    int i = blockIdx.x * blockDim.x + threadIdx.x;
    int stride = gridDim.x * blockDim.x;
    for (int j = i; j < na; j += stride) a[j] = 0.0f;
    for (int j = i; j < nb; j += stride) b[j] = 0.0f;
}

// ---------------- kernel 2: degree counting (float atomics) ----------------
__global__ void degree_k(const int* __restrict__ src, const int* __restrict__ dst,
                         float* __restrict__ out_cnt, float* __restrict__ in_cnt) {
    int e = blockIdx.x * blockDim.x + threadIdx.x;
    if (e < N_EDGES) {
        atomicAdd(&out_cnt[src[e]], 1.0f);
        atomicAdd(&in_cnt[dst[e]], 1.0f);
    }
}

// ---------------- kernel 2b: pre-swizzle W into per-lane WMMA B-fragment order --------
// WB[kk*32 + lane] = float4{ B0.x, B0.y, B1.x, B1.y } for V_WMMA_F32_16X16X4_F32:
//   lane -> N = lane&15 (panel adds +16), VGPR pair -> K rows kb, kb+1,
//   kb = 4*kk + 2*(lane>>4).  Each W element appears exactly once (32 KB total).
__global__ void pack_w_k(const float* __restrict__ W, float* __restrict__ WB) {
    int idx = blockIdx.x * blockDim.x + threadIdx.x;
    if (idx < KSTEPS * 32) {
        int kk = idx >> 5, l = idx & 31;
        int kb = kk * 4 + ((l >> 4) << 1);
        int n  = l & 15;
        v4f o;
        o.x = W[(size_t)kb       * OUT_FEATS + n];
        o.y = W[(size_t)(kb + 1) * OUT_FEATS + n];
        o.z = W[(size_t)kb       * OUT_FEATS + 16 + n];
        o.w = W[(size_t)(kb + 1) * OUT_FEATS + 16 + n];
        ((v4f*)WB)[idx] = o;
    }
}

// ---------------- kernel 3: scatter top-k -> transposed LDS tile -> f32 WMMA ----------
// Block = 128 threads = 4 waves; each wave computes y for 16 nodes.
// A tile stored transposed in LDS: tileT[k][m] (16 floats per k-row) ->
//   lanes 0-15 read banks kb*16+0..15, lanes 16-31 read +32..+47: conflict-free.
__global__ __launch_bounds__(128) void maxk_gemm_k(
    const float* __restrict__ topk_vals, const int* __restrict__ topk_idx,
    const float* __restrict__ WB,         // swizzled fragments [64][32] float4
    const float* __restrict__ out_cnt,
    float* __restrict__ y)                // [N][32]
{
    __shared__ float tileT[4 * IN_FEATS * 16];   // 4 waves x 16 KB = 64 KB

    const int t    = threadIdx.x;
    const int wave = t >> 5;
    const int lane = t & 31;

    // zero all tiles (16384 floats, vectorized)
    for (int i = t; i < (4 * IN_FEATS * 16) / 4; i += 128)
        ((v4f*)tileT)[i] = (v4f){0.f, 0.f, 0.f, 0.f};
    __syncthreads();

    const int nodeBase = blockIdx.x * 64 + wave * 16;
    float* tw = &tileT[wave * IN_FEATS * 16];

    // lanes 0..15: each owns one node = one column of tileT; serial k-loop so
    // duplicate indices accumulate exactly like jnp .at[].add
    if (lane < 16) {
        int n = nodeBase + lane;
        if (n < N_NODES) {
            float s = 1.0f / sqrtf(fmaxf(out_cnt[n], 1.0f));
            const float* v  = topk_vals + (size_t)n * K_TOP;
            const int*   ix = topk_idx  + (size_t)n * K_TOP;
            #pragma unroll 4
            for (int k = 0; k < K_TOP; ++k)
                tw[ix[k] * 16 + lane] += v[k] * s;
        }
    }
    __syncthreads();

    // A frag (16x4 f32): lane -> M = lane&15; VGPR pair -> K = 4*kk + 2*(lane>>4) + {0,1}
    v8f c0 = {}; v8f c1 = {};
    const int m     = lane & 15;
    const int khalf = (lane >> 4) * 2;
    const v4f* __restrict__ WBv = (const v4f*)WB;

    #pragma unroll 8
    for (int kk = 0; kk < KSTEPS; ++kk) {
        const int kb = kk * 4 + khalf;
        v2f a;
        a.x = tw[kb * 16 + m];
        a.y = tw[(kb + 1) * 16 + m];
        v4f b  = WBv[kk * 32 + lane];          // one coalesced b128 per k-step
        v2f b0 = {b.x, b.y};
        v2f b1 = {b.z, b.w};
        c0 = __builtin_amdgcn_wmma_f32_16x16x4_f32(false, a, false, b0, (short)0, c0, false, false);
        c1 = __builtin_amdgcn_wmma_f32_16x16x4_f32(false, a, false, b1, (short)0, c1, false, false);
    }

    // D layout: VGPR r -> M = r + 8*(lane>>4), N = lane&15
    const int mh = (lane >> 4) * 8;
    if (nodeBase + 16 <= N_NODES) {            // wave-uniform fast path, branch-free stores
        float* yb = y + (size_t)nodeBase * OUT_FEATS;
        #pragma unroll
        for (int r = 0; r < 8; ++r) {
            int n = mh + r;
            yb[n * OUT_FEATS + m]      = c0[r];
            yb[n * OUT_FEATS + 16 + m] = c1[r];
        }
    } else {
        #pragma unroll
        for (int r = 0; r < 8; ++r) {
            int n = nodeBase + mh + r;
            if (n < N_NODES) {
                y[(size_t)n * OUT_FEATS + m]      = c0[r];
                y[(size_t)n * OUT_FEATS + 16 + m] = c1[r];
            }
        }
    }
}

// ---------------- kernel 4: SpMM aggregation, edge x channel parallel ----------------
__global__ void spmm_k(const int* __restrict__ src, const int* __restrict__ dst,
                       const float* __restrict__ y, float* __restrict__ out) {
    int gid = blockIdx.x * blockDim.x + threadIdx.x;   // E*32 = 51.2M < 2^31
    if (gid < N_EDGES * OUT_FEATS) {
        int e  = gid >> 5;
        int ch = gid & 31;
        int s = src[e], d = dst[e];
        atomicAdd(&out[(size_t)d * OUT_FEATS + ch], y[(size_t)s * OUT_FEATS + ch]);
    }
}

// ---------------- kernel 5: right-normalize + bias ----------------
__global__ void finalize_k(float* __restrict__ out, const float* __restrict__ in_cnt,
                           const float* __restrict__ bias) {
    int i = blockIdx.x * blockDim.x + threadIdx.x;
    if (i < N_NODES * OUT_FEATS) {
        int n  = i >> 5;
        int ch = i & 31;
        out[i] = out[i] * (1.0f / sqrtf(fmaxf(in_cnt[n], 1.0f))) + bias[ch];
    }
}

extern "C" void kernel_launch(void* const* d_in, const int* in_sizes, int n_in,
                              void* d_out, int out_size, void* d_ws, size_t ws_size,
                              hipStream_t stream) {
    const float* topk_vals = (const float*)d_in[0];
    const int*   topk_idx  = (const int*)  d_in[1];
    const int*   src       = (const int*)  d_in[2];
    const int*   dst       = (const int*)  d_in[3];
    const float* W         = (const float*)d_in[4];
    const float* bias      = (const float*)d_in[5];
    float*       out       = (float*)d_out;

    // workspace layout (floats): out_cnt[N] | in_cnt[N] | y[N*32] | WB[8192]
    float* ws      = (float*)d_ws;
    float* out_cnt = ws;
    float* in_cnt  = ws + N_NODES;
    float* y       = ws + 2 * N_NODES;
    float* WB      = ws + 2 * N_NODES + N_NODES * OUT_FEATS;

    // 1) zero counters + output accumulator
    zero_k<<<2048, 256, 0, stream>>>(out_cnt, 2 * N_NODES, out, N_NODES * OUT_FEATS);

    // 2) degrees
    degree_k<<<(N_EDGES + 255) / 256, 256, 0, stream>>>(src, dst, out_cnt, in_cnt);

    // 2b) swizzle W into WMMA B-fragment order
    pack_w_k<<<(KSTEPS * 32 + 255) / 256, 256, 0, stream>>>(W, WB);

    // 3) left-normalized sparse-to-dense + WMMA GEMM -> y
    maxk_gemm_k<<<(N_NODES + 63) / 64, 128, 0, stream>>>(topk_vals, topk_idx, WB, out_cnt, y);

    // 4) aggregation: out[d] += y[s] over edges
    spmm_k<<<(N_EDGES * OUT_FEATS + 255) / 256, 256, 0, stream>>>(src, dst, y, out);

    // 5) right normalization + bias
    finalize_k<<<(N_NODES * OUT_FEATS + 255) / 256, 256, 0, stream>>>(out, in_cnt, bias);
}